// Cross_VQ_RA_2937757630652
// MI455X (gfx1250) — compile-verified
//
#include <hip/hip_runtime.h>
#include <math.h>

typedef __attribute__((ext_vector_type(2))) float v2f;
typedef __attribute__((ext_vector_type(8))) float v8f;

#define N_ROWS 8192
#define K_CB   8192
#define DIM    512
#define TILE_ROWS 128
#define BLOCK 256
// 128-row A tile (256 KB) + 32-row B tile (64 KB) = 320 KB = CDNA5 WGP LDS max
#define LDS_FLOATS (TILE_ROWS*DIM + 32*DIM)

// ---------------------------------------------------------------------------
// Zero workspace accumulators (ws is poisoned 0xAA; must re-init every call)
// ---------------------------------------------------------------------------
__global__ void k_zero(float* __restrict__ hist, double* __restrict__ sums) {
    int g = blockIdx.x * blockDim.x + threadIdx.x;
    if (g < 2 * K_CB) hist[g] = 0.0f;
    if (g < 4) sums[g] = 0.0;
}

// ---------------------------------------------------------------------------
// Kernel 1: QC[k][d] = sum_j E[k][j] * W[d][j] + b[d]   (fp32 WMMA 16x16x4)
// 8 waves * 16-row tiles; dual 16-col chunks per staged 32-row W tile.
// ---------------------------------------------------------------------------
__global__ __launch_bounds__(BLOCK)
void k_project(const float* __restrict__ E, const float* __restrict__ W,
               const float* __restrict__ b, float* __restrict__ QC) {
    extern __shared__ float lds[];
    float* XL = lds;                    // TILE_ROWS x DIM
    float* BL = lds + TILE_ROWS * DIM;  // 32 x DIM
    const int tid  = threadIdx.x;
    const int row0 = blockIdx.x * TILE_ROWS;

    for (int i = tid * 4; i < TILE_ROWS * DIM; i += BLOCK * 4)
        *(float4*)&XL[i] = *(const float4*)&E[(size_t)row0 * DIM + i];

    const int wave = tid >> 5, lane = tid & 31;
    const int half = lane >> 4, l15 = lane & 15;
    const float* Arow  = &XL[(wave * 16 + l15) * DIM + 2 * half];
    const float* Brow0 = &BL[l15 * DIM + 2 * half];
    const float* Brow1 = &BL[(16 + l15) * DIM + 2 * half];

    for (int c0 = 0; c0 < DIM; c0 += 32) {
        __syncthreads();  // also covers initial XL stage
        for (int i = tid * 4; i < 32 * DIM; i += BLOCK * 4)
            *(float4*)&BL[i] = *(const float4*)&W[(size_t)c0 * DIM + i];
        __syncthreads();

        float bv0 = b[c0 + l15];
        float bv1 = b[c0 + 16 + l15];
        v8f acc0 = {bv0, bv0, bv0, bv0, bv0, bv0, bv0, bv0};
        v8f acc1 = {bv1, bv1, bv1, bv1, bv1, bv1, bv1, bv1};
        #pragma unroll 8
        for (int k = 0; k < DIM; k += 4) {
            v2f a  = *(const v2f*)&Arow[k];
            v2f b0 = *(const v2f*)&Brow0[k];
            v2f b1 = *(const v2f*)&Brow1[k];
            acc0 = __builtin_amdgcn_wmma_f32_16x16x4_f32(
                false, a, false, b0, (short)0, acc0, false, false);
            acc1 = __builtin_amdgcn_wmma_f32_16x16x4_f32(
                false, a, false, b1, (short)0, acc1, false, false);
        }
        #pragma unroll
        for (int v = 0; v < 8; ++v) {
            int r = row0 + wave * 16 + v + 8 * half;
            QC[(size_t)r * DIM + c0 + l15]      = acc0[v];
            QC[(size_t)r * DIM + c0 + 16 + l15] = acc1[v];
        }
    }
}

// ---------------------------------------------------------------------------
// Kernel 1b: cb_sq[k] = ||QC[k]||^2
// ---------------------------------------------------------------------------
__global__ __launch_bounds__(BLOCK)
void k_cbsq(const float* __restrict__ QC, float* __restrict__ cbsq) {
    int r = blockIdx.x * BLOCK + threadIdx.x;
    const float4* p = (const float4*)(QC + (size_t)r * DIM);
    float s = 0.0f;
    #pragma unroll 4
    for (int j = 0; j < DIM / 4; ++j) {
        float4 v = p[j];
        s += v.x * v.x + v.y * v.y + v.z * v.z + v.w * v.w;
    }
    cbsq[r] = s;
}

// ---------------------------------------------------------------------------
// Kernel 2: per-row argmin over K of (cb_sq[k] - 2 x.QC[k]) via fp32 WMMA.
// grid.y selects stream (0 = scRNA, 1 = ribo). Dual chunks per staged tile:
// A fragment reused for 2 wmma; two independent accumulation chains.
// ---------------------------------------------------------------------------
__global__ __launch_bounds__(BLOCK)
void k_nearest(const float* __restrict__ X0, const float* __restrict__ X1,
               const float* __restrict__ QC, const float* __restrict__ cbsq,
               int* __restrict__ idxAll) {
    extern __shared__ float lds[];
    float* XL = lds;
    float* CL = lds + TILE_ROWS * DIM;  // 32 codebook rows
    const float* X = blockIdx.y ? X1 : X0;
    int* idx_out = idxAll + blockIdx.y * N_ROWS;

    const int tid  = threadIdx.x;
    const int row0 = blockIdx.x * TILE_ROWS;
    for (int i = tid * 4; i < TILE_ROWS * DIM; i += BLOCK * 4)
        *(float4*)&XL[i] = *(const float4*)&X[(size_t)row0 * DIM + i];

    const int wave = tid >> 5, lane = tid & 31;
    const int half = lane >> 4, l15 = lane & 15;
    const float* Arow  = &XL[(wave * 16 + l15) * DIM + 2 * half];
    const float* Brow0 = &CL[l15 * DIM + 2 * half];
    const float* Brow1 = &CL[(16 + l15) * DIM + 2 * half];

    float minv[8]; int mini[8];
    #pragma unroll
    for (int v = 0; v < 8; ++v) { minv[v] = 3.4e38f; mini[v] = 0; }

    for (int c0 = 0; c0 < K_CB; c0 += 32) {
        __syncthreads();  // previous chunk fully consumed (also covers XL stage)
        for (int i = tid * 4; i < 32 * DIM; i += BLOCK * 4)
            *(float4*)&CL[i] = *(const float4*)&QC[(size_t)c0 * DIM + i];
        __syncthreads();

        v8f acc0 = {0.f, 0.f, 0.f, 0.f, 0.f, 0.f, 0.f, 0.f};
        v8f acc1 = {0.f, 0.f, 0.f, 0.f, 0.f, 0.f, 0.f, 0.f};
        #pragma unroll 8
        for (int k = 0; k < DIM; k += 4) {
            v2f a  = *(const v2f*)&Arow[k];
            v2f b0 = *(const v2f*)&Brow0[k];
            v2f b1 = *(const v2f*)&Brow1[k];
            acc0 = __builtin_amdgcn_wmma_f32_16x16x4_f32(
                false, a, false, b0, (short)0, acc0, false, false);
            acc1 = __builtin_amdgcn_wmma_f32_16x16x4_f32(
                false, a, false, b1, (short)0, acc1, false, false);
        }
        const int col0 = c0 + l15;
        const int col1 = c0 + 16 + l15;
        const float cs0 = cbsq[col0];
        const float cs1 = cbsq[col1];
        #pragma unroll
        for (int v = 0; v < 8; ++v) {
            float d0 = cs0 - 2.0f * acc0[v];
            if (d0 < minv[v]) { minv[v] = d0; mini[v] = col0; }
            float d1 = cs1 - 2.0f * acc1[v];
            if (d1 < minv[v]) { minv[v] = d1; mini[v] = col1; }
        }
    }

    // argmin across the 16 lanes of each half (cols); tie -> smaller index
    #pragma unroll
    for (int m = 8; m >= 1; m >>= 1) {
        #pragma unroll
        for (int v = 0; v < 8; ++v) {
            float ov = __shfl_xor(minv[v], m, 32);
            int   oi = __shfl_xor(mini[v], m, 32);
            if (ov < minv[v] || (ov == minv[v] && oi < mini[v])) {
                minv[v] = ov; mini[v] = oi;
            }
        }
    }
    if (l15 == 0) {
        #pragma unroll
        for (int v = 0; v < 8; ++v)
            idx_out[row0 + wave * 16 + v + 8 * half] = mini[v];
    }
}

// ---------------------------------------------------------------------------
// Kernel 3: gather quantized rows to output + accumulate 4 squared-error sums
// sums[0]=S(scR-qR)^2 sums[1]=S(rib-qB)^2 sums[2]=S(qB-scR)^2 sums[3]=S(qR-rib)^2
// ---------------------------------------------------------------------------
__global__ __launch_bounds__(BLOCK)
void k_gather(const float* __restrict__ xR, const float* __restrict__ xB,
              const float* __restrict__ QC, const int* __restrict__ idxAll,
              float* __restrict__ outR, float* __restrict__ outB,
              double* __restrict__ sums) {
    __shared__ float red[BLOCK];
    const int i = blockIdx.x;
    const int tid = threadIdx.x;
    const int iR = idxAll[i];
    const int iB = idxAll[N_ROWS + i];
    const float* qRp = QC + (size_t)iR * DIM;
    const float* qBp = QC + (size_t)iB * DIM;

    float e0 = 0.f, e1 = 0.f, e2 = 0.f, e3 = 0.f;
    for (int j = tid; j < DIM; j += BLOCK) {
        float xr = xR[(size_t)i * DIM + j];
        float xb = xB[(size_t)i * DIM + j];
        float qr = qRp[j], qb = qBp[j];
        outR[(size_t)i * DIM + j] = qr;
        outB[(size_t)i * DIM + j] = qb;
        float d0 = xr - qr, d1 = xb - qb, d2 = qb - xr, d3 = qr - xb;
        e0 += d0 * d0; e1 += d1 * d1; e2 += d2 * d2; e3 += d3 * d3;
    }
    float vals[4] = {e0, e1, e2, e3};
    for (int t = 0; t < 4; ++t) {
        red[tid] = vals[t]; __syncthreads();
        for (int s = BLOCK / 2; s > 0; s >>= 1) {
            if (tid < s) red[tid] += red[tid + s];
            __syncthreads();
        }
        if (tid == 0) atomicAdd(&sums[t], (double)red[0]);
        __syncthreads();
    }
}

// ---------------------------------------------------------------------------
// Kernel 4: histograms of indices (counts as f32; exact up to 2^24)
// ---------------------------------------------------------------------------
__global__ void k_hist(const int* __restrict__ idxAll, float* __restrict__ hist) {
    int g = blockIdx.x * blockDim.x + threadIdx.x;
    if (g < 2 * N_ROWS) {
        int s = g >> 13;  // N_ROWS == 8192 == 2^13
        atomicAdd(&hist[s * K_CB + idxAll[g]], 1.0f);
    }
}

// ---------------------------------------------------------------------------
// Kernel 5: losses + perplexities -> d_out scalars
// ---------------------------------------------------------------------------
__global__ __launch_bounds__(BLOCK)
void k_final(const float* __restrict__ hist, const double* __restrict__ sums,
             float* __restrict__ outS) {
    __shared__ float red[BLOCK];
    const int tid = threadIdx.x;
    float perp[2];
    for (int s = 0; s < 2; ++s) {
        float acc = 0.f;
        for (int g = tid; g < K_CB; g += BLOCK) {
            float p = hist[s * K_CB + g] * (1.0f / (float)N_ROWS);
            acc += p * logf(p + 1e-10f);
        }
        red[tid] = acc; __syncthreads();
        for (int st = BLOCK / 2; st > 0; st >>= 1) {
            if (tid < st) red[tid] += red[tid + st];
            __syncthreads();
        }
        perp[s] = expf(-red[0]);
        __syncthreads();
    }
    if (tid == 0) {
        const double inv = 1.0 / ((double)N_ROWS * (double)DIM);
        double m0 = sums[0] * inv;  // mean (scRNA - scRNA_q)^2
        double m1 = sums[1] * inv;  // mean (ribo  - ribo_q )^2
        double m2 = sums[2] * inv;  // mean (ribo_q  - scRNA)^2
        double m3 = sums[3] * inv;  // mean (scRNA_q - ribo )^2
        outS[0] = (float)(0.5 * m0);                                        // scRNA_loss
        outS[1] = (float)(0.75 * m1 + 0.25 * m0 + 0.125 * m2 + 0.125 * m3); // ribo_loss
        outS[2] = perp[0];
        outS[3] = perp[1];
    }
}

// ---------------------------------------------------------------------------
extern "C" void kernel_launch(void* const* d_in, const int* in_sizes, int n_in,
                              void* d_out, int out_size, void* d_ws, size_t ws_size,
                              hipStream_t stream) {
    (void)in_sizes; (void)n_in; (void)out_size; (void)ws_size;
    const float* scRNA = (const float*)d_in[0];
    const float* ribo  = (const float*)d_in[1];
    const float* emb   = (const float*)d_in[2];
    const float* projw = (const float*)d_in[3];
    const float* projb = (const float*)d_in[4];
    // d_in[5] == flag (unused by the forward math)

    char* ws = (char*)d_ws;
    float*  QC   = (float*)ws;                                        // K*D f32
    float*  cbsq = (float*)(ws + (size_t)K_CB * DIM * 4);             // K f32
    int*    idxA = (int*)  (ws + (size_t)K_CB * DIM * 4 + K_CB * 4);  // 2*N i32
    float*  hist = (float*)((char*)idxA + (size_t)2 * N_ROWS * 4);    // 2*K f32
    double* sums = (double*)((char*)hist + (size_t)2 * K_CB * 4);     // 4 f64

    float* outR = (float*)d_out;
    float* outB = outR + (size_t)N_ROWS * DIM;
    float* outS = outB + (size_t)N_ROWS * DIM;

    const size_t ldsBytes = (size_t)LDS_FLOATS * sizeof(float);  // 320 KB

    hipLaunchKernelGGL(k_zero,    dim3(64),                 dim3(256),  0,        stream, hist, sums);
    hipLaunchKernelGGL(k_project, dim3(K_CB / TILE_ROWS),   dim3(BLOCK), ldsBytes, stream, emb, projw, projb, QC);
    hipLaunchKernelGGL(k_cbsq,    dim3(K_CB / BLOCK),       dim3(BLOCK), 0,        stream, QC, cbsq);
    hipLaunchKernelGGL(k_nearest, dim3(N_ROWS / TILE_ROWS, 2), dim3(BLOCK), ldsBytes, stream, scRNA, ribo, QC, cbsq, idxA);
    hipLaunchKernelGGL(k_gather,  dim3(N_ROWS),             dim3(BLOCK), 0,        stream, scRNA, ribo, QC, idxA, outR, outB, sums);
    hipLaunchKernelGGL(k_hist,    dim3((2 * N_ROWS) / 256), dim3(256),  0,        stream, idxA, hist);
    hipLaunchKernelGGL(k_final,   dim3(1),                  dim3(BLOCK), 0,        stream, hist, sums, outS);
}